// BetaBernoulliMixture_712964571408
// MI455X (gfx1250) — compile-verified
//
#include <hip/hip_runtime.h>
#include <stdint.h>

// Problem constants (fixed by the reference).
#define BSZ 4096
#define TSZ 8192
#define CHUNK 256                 // one 16x16 f32 tile per scan step
#define NCHUNK (TSZ / CHUNK)      // 32 chunks per row
#define NBUF 4                    // async pipeline depth (3 chunks in flight)
#define BT ((size_t)BSZ * (size_t)TSZ)

typedef float v2f __attribute__((ext_vector_type(2)));
typedef float v8f __attribute__((ext_vector_type(8)));

// -------- CDNA5 async global->LDS staging (ASYNCcnt-tracked) --------------
__device__ __forceinline__ uint32_t lds_off32(const void* p) {
  // LDS aperture: flat addr low 32 bits == LDS byte offset (ISA 10.2).
  return (uint32_t)(uintptr_t)p;
}

__device__ __forceinline__ void async_load_chunk(const float* gsrc, uint32_t lds_base, int lane) {
  // 256 floats = 1024B = two b128 async loads (16B/lane * 32 lanes each).
  uint64_t ga = (uint64_t)(uintptr_t)gsrc + (uint64_t)lane * 16u;
  uint32_t la = lds_base + (uint32_t)lane * 16u;
  asm volatile("global_load_async_to_lds_b128 %0, %1, off" :: "v"(la), "v"(ga) : "memory");
  uint64_t ga2 = ga + 512u;
  uint32_t la2 = la + 512u;
  asm volatile("global_load_async_to_lds_b128 %0, %1, off" :: "v"(la2), "v"(ga2) : "memory");
}

#define S_WAIT_ASYNCCNT(n) asm volatile("s_wait_asynccnt %0" :: "n"(n) : "memory")

// -------- WMMA-based 16x16 exclusive prefix scan --------------------------
// D = A(16x4) x B(4x16) + C chained over 4 K-slices.
__device__ __forceinline__ v8f wmma_chain(const v2f a[4], const v2f b[4]) {
  v8f acc = {};
#pragma unroll
  for (int c = 0; c < 4; ++c) {
    acc = __builtin_amdgcn_wmma_f32_16x16x4_f32(false, a[c], false, b[c],
                                                (short)0, acc, false, false);
  }
  return acc;
}

// x: 16x16 row-major tile in LDS (element t_local = m*16 + k).
// Returns, in C/D layout (VGPR r <-> rows r and r+8), the *exclusive* prefix
// over the full 256-element chunk (row scan + cross-row carries), and the
// chunk total (wave-uniform) via *total_out.
// tri[c] holds [k0 < lane%16] masks: serves as B = U_strict for the row scan
// and as A = L_strict for the carry matmul (per-lane values coincide).
__device__ __forceinline__ v8f chunk_exclusive_scan(const float* __restrict__ x,
                                                    float* __restrict__ rbuf,
                                                    const v2f tri[4],
                                                    int row, int hi,
                                                    float* total_out) {
  // A = data tile in A-layout: lane (row=L%16) reads column pair (4c+2*hi, +1).
  v2f a[4];
#pragma unroll
  for (int c = 0; c < 4; ++c) {
    int k0 = 4 * c + 2 * hi;
    a[c] = *(const v2f*)(x + row * 16 + k0);
  }
  v8f E = wmma_chain(a, tri);            // E[m][n] = sum_{k<n} X[m][k]

  // Row totals R[m] (lanes 0..15 each sum their row from LDS).
  {
    const float4* xr = (const float4*)(x + row * 16);
    float4 s0 = xr[0], s1 = xr[1], s2 = xr[2], s3 = xr[3];
    float rs = (s0.x + s0.y + s0.z + s0.w) + (s1.x + s1.y + s1.z + s1.w) +
               (s2.x + s2.y + s2.z + s2.w) + (s3.x + s3.y + s3.z + s3.w);
    if (hi == 0) rbuf[row] = rs;         // same-wave DS ops are in-order
  }

  // B2[k][n] = R[k] broadcast across n: lane reads pair (R[k0], R[k0+1]).
  v2f b2[4];
#pragma unroll
  for (int c = 0; c < 4; ++c) {
    int k0 = 4 * c + 2 * hi;
    b2[c] = *(const v2f*)(rbuf + k0);
  }
  v8f C2 = wmma_chain(tri, b2);          // C2[m][*] = sum_{j<m} R[j]

  // Wave-uniform chunk total.
  const float4* rv = (const float4*)rbuf;
  float4 t0 = rv[0], t1 = rv[1], t2 = rv[2], t3 = rv[3];
  *total_out = (t0.x + t0.y + t0.z + t0.w) + (t1.x + t1.y + t1.z + t1.w) +
               (t2.x + t2.y + t2.z + t2.w) + (t3.x + t3.y + t3.z + t3.w);
  return E + C2;                         // exclusive prefix within chunk
}

// ---------------------------------------------------------------------------
__global__ __launch_bounds__(32) void BetaBernoulliMixture_kernel(
    const float* __restrict__ obs,
    const float* __restrict__ A1, const float* __restrict__ B1,
    const float* __restrict__ A2, const float* __restrict__ B2,
    const float* __restrict__ MW,
    float* __restrict__ out) {
  __shared__ float lds_obs[NBUF][CHUNK]; // 4-deep async staging pipeline
  __shared__ float lds_d[CHUNK];         // transpose bounce for log-term scan
  __shared__ float lds_R[16];
  __shared__ float lds_Rd[16];

  const int b = blockIdx.x;              // one wave32 per row
  const int lane = threadIdx.x;
  const int hi = lane >> 4;
  const int n = lane & 15;               // column index (B/C/D layouts), row index (A layout)

  const float a1 = A1[b], b1 = B1[b], a2 = A2[b], b2 = B2[b];
  const float mw = MW[0];
  const float omw = 1.0f - mw;
  const float s1 = a1 + b1, s2 = a2 + b2;

  // Strict-triangular 0/1 constants (shared by U_strict-as-B and L_strict-as-A).
  v2f tri[4];
#pragma unroll
  for (int c = 0; c < 4; ++c) {
    int k0 = 4 * c + 2 * hi;
    tri[c].x = (k0 < n) ? 1.0f : 0.0f;
    tri[c].y = (k0 + 1 < n) ? 1.0f : 0.0f;
  }

  const float* row_obs = obs + (size_t)b * TSZ;
  float* o0 = out + 0 * BT + (size_t)b * TSZ;  // a1
  float* o1 = out + 1 * BT + (size_t)b * TSZ;  // b1
  float* o2 = out + 2 * BT + (size_t)b * TSZ;  // a2
  float* o3 = out + 3 * BT + (size_t)b * TSZ;  // b2
  float* o4 = out + 4 * BT + (size_t)b * TSZ;  // post_mixweight

  float carry_s = 0.0f;   // prefix of successes across chunks
  float carry_d = 0.0f;   // prefix of log2-ratio terms across chunks

  // Prologue: fill the async pipeline 3 chunks deep (6 ASYNCcnt entries).
#pragma unroll
  for (int p = 0; p < NBUF - 1; ++p)
    async_load_chunk(row_obs + (size_t)p * CHUNK, lds_off32(&lds_obs[p][0]), lane);

  for (int it = 0; it < NCHUNK; ++it) {
    const int buf = it & (NBUF - 1);
    if (it + (NBUF - 1) < NCHUNK) {
      async_load_chunk(row_obs + (size_t)(it + NBUF - 1) * CHUNK,
                       lds_off32(&lds_obs[(it + NBUF - 1) & (NBUF - 1)][0]), lane);
      S_WAIT_ASYNCCNT(6);                // 3 younger chunks outstanding
    } else {
      const int rem = NCHUNK - 1 - it;   // younger chunks still in flight
      if (rem == 2)      S_WAIT_ASYNCCNT(4);
      else if (rem == 1) S_WAIT_ASYNCCNT(2);
      else               S_WAIT_ASYNCCNT(0);
    }

    const float* xb = &lds_obs[buf][0];
    const int t0 = it * CHUNK;

    // --- scan #1: exclusive prefix of obs -> s_prev -----------------------
    float totS;
    v8f sprev = chunk_exclusive_scan(xb, lds_R, tri, n, hi, &totS);

    // --- per-element log2-ratio term (2x v_log_f32, no divide) ------------
    float spv[8], fpv[8];
#pragma unroll
    for (int r = 0; r < 8; ++r) {
      int m = r + hi * 8;
      int tloc = m * 16 + n;             // D-layout element index in chunk
      float o = xb[tloc];
      float sp = sprev[r] + carry_s;     // successes strictly before t
      float tf = (float)(t0 + tloc);     // s_prev + f_prev = t (exact in f32)
      float fp = tf - sp;
      bool one = (o > 0.5f);
      float num2 = one ? (a2 + sp) : (b2 + fp);
      float num1 = one ? (a1 + sp) : (b1 + fp);
      // log2 domain: d = log2(num2*(s1+t)) - log2(num1*(s2+t))
      float d = __builtin_amdgcn_logf(num2 * (s1 + tf)) -
                __builtin_amdgcn_logf(num1 * (s2 + tf));
      lds_d[tloc] = d;
      spv[r] = sp;
      fpv[r] = fp;
    }

    // --- scan #2: exclusive prefix of d -> log2(c2ratio) ------------------
    float totD;
    v8f dscan = chunk_exclusive_scan(lds_d, lds_Rd, tri, n, hi, &totD);

    // --- outputs (write-once streams: non-temporal) -----------------------
#pragma unroll
    for (int r = 0; r < 8; ++r) {
      int m = r + hi * 8;
      size_t t = (size_t)(t0 + m * 16 + n);
      float cl = dscan[r] + carry_d;     // log2(exp(clog2-clog1))
      // pw = mw / (mw + (1-mw)*2^cl): one v_exp + one v_rcp, no div ladder.
      float denom = mw + omw * __builtin_amdgcn_exp2f(cl);
      float pw = mw * __builtin_amdgcn_rcpf(denom);
      __builtin_nontemporal_store(a1 + spv[r], o0 + t);
      __builtin_nontemporal_store(b1 + fpv[r], o1 + t);
      __builtin_nontemporal_store(a2 + spv[r], o2 + t);
      __builtin_nontemporal_store(b2 + fpv[r], o3 + t);
      __builtin_nontemporal_store(pw, o4 + t);
    }

    carry_s += totS;
    carry_d += totD;
  }
}

extern "C" void kernel_launch(void* const* d_in, const int* in_sizes, int n_in,
                              void* d_out, int out_size, void* d_ws, size_t ws_size,
                              hipStream_t stream) {
  const float* obs = (const float*)d_in[0];
  const float* a1  = (const float*)d_in[1];
  const float* b1  = (const float*)d_in[2];
  const float* a2  = (const float*)d_in[3];
  const float* b2  = (const float*)d_in[4];
  const float* mw  = (const float*)d_in[5];
  (void)in_sizes; (void)n_in; (void)out_size; (void)d_ws; (void)ws_size;
  BetaBernoulliMixture_kernel<<<dim3(BSZ), dim3(32), 0, stream>>>(
      obs, a1, b1, a2, b2, mw, (float*)d_out);
}